// bfan_ed_6365141532897
// MI455X (gfx1250) — compile-verified
//
#include <hip/hip_runtime.h>
#include <math.h>

// ---------------- Geometry (fixed by the reference) ----------------
#define VIEWS 512
#define DETS  736
#define IMG_W 256
#define IMG_H 256
#define BATCH 4

constexpr float D_IMG  = 0.006641f;
constexpr float S2R    = 0.5944f;
constexpr float VIRDET = (float)(0.0072 * 0.5944 / (0.5944 + 0.4904));
constexpr float D_ANG  = (float)(2.0 * 3.14159265358979323846 / 512.0);
constexpr float COEF   = (float)(3.14159265358979323846 / 512.0);

typedef __attribute__((ext_vector_type(2))) float v2f;
typedef __attribute__((ext_vector_type(8))) float v8f;

// -------------------------------------------------------------------
// Kernel 1: build the fused filter matrix (transposed) + trig table.
//   Ft[d][k] = COEF * VIRDET * w[k] * filt[d - k + DETS-1]
// Folds cosine weighting (input-side) and the final pi/views scale
// into the GEMM operand, so the rest of the pipeline is pure matmul
// + weighted gather-accumulate.
// -------------------------------------------------------------------
__global__ void fbp_build_tables(const float* __restrict__ filt,
                                 const float* __restrict__ w,
                                 float* __restrict__ Ft,
                                 float* __restrict__ trig) {
    const int gid = blockIdx.x * blockDim.x + threadIdx.x;
    const int total = DETS * DETS;
    if (gid < total) {
        const int d = gid / DETS;
        const int k = gid - d * DETS;
        Ft[gid] = (COEF * VIRDET) * w[k] * filt[d - k + (DETS - 1)];
    }
    if (gid < VIEWS) {
        const float beta = (float)gid * D_ANG;
        trig[2 * gid + 0] = cosf(beta);
        trig[2 * gid + 1] = sinf(beta);
    }
}

// -------------------------------------------------------------------
// Kernel 2: ramp-filter GEMM on the fp32 matrix cores.
//   Q[n,d] = sum_k P[n,k] * Ft[d,k]     n in [0,2048), d,k in [0,736)
// One wave owns a 64x16 strip of Q (4 M-tiles x 1 N-tile), reusing the
// B operand across the 4 WMMAs per K-step. K advances 4 per WMMA
// (V_WMMA_F32_16X16X4_F32), 184 steps.
//
// Operand layouts (wave32, per ISA 7.12.2):
//   A 16x4 : lane l (lo=l&15, hi=l>>4): v0 = A[lo][2*hi], v1 = A[lo][2*hi+1]
//   B 4x16 : v0 = B[2*hi][lo], v1 = B[2*hi+1][lo]
//   C 16x16: vr = C[r + 8*hi][lo]
// Ft is stored [d][k] so both A and B lane loads are contiguous float2.
// -------------------------------------------------------------------
__global__ void __launch_bounds__(256)
fbp_filter_gemm(const float* __restrict__ P,    // [2048, 736] row-major
                const float* __restrict__ Ft,   // [736, 736]  Ft[d][k]
                float* __restrict__ Q) {        // [2048, 736] row-major
    const int lane = threadIdx.x & 31;
    const int wave = threadIdx.x >> 5;
    const int lo   = lane & 15;
    const int hi   = lane >> 4;                 // 0 or 1

    const int tileN = blockIdx.x;                       // 0..45
    const int m0    = (blockIdx.y * 8 + wave) * 4;      // first of 4 M-tiles (0..124)
    const int col   = tileN * 16 + lo;

    const float* brow  = Ft + (size_t)col * DETS;                 // B[k][col]
    const float* arow0 = P  + (size_t)(m0 * 16 + lo) * DETS;      // A rows, tile 0
    const float* arow1 = arow0 + (size_t)16 * DETS;
    const float* arow2 = arow0 + (size_t)32 * DETS;
    const float* arow3 = arow0 + (size_t)48 * DETS;

    v8f acc0 = {}, acc1 = {}, acc2 = {}, acc3 = {};

    const int koff = 2 * hi;
#pragma unroll 4
    for (int k = 0; k < DETS; k += 4) {
        const int kk = k + koff;
        const v2f bv = *(const v2f*)(brow  + kk);
        const v2f a0 = *(const v2f*)(arow0 + kk);
        const v2f a1 = *(const v2f*)(arow1 + kk);
        const v2f a2 = *(const v2f*)(arow2 + kk);
        const v2f a3 = *(const v2f*)(arow3 + kk);
        acc0 = __builtin_amdgcn_wmma_f32_16x16x4_f32(false, a0, false, bv, (short)0, acc0, false, false);
        acc1 = __builtin_amdgcn_wmma_f32_16x16x4_f32(false, a1, false, bv, (short)0, acc1, false, false);
        acc2 = __builtin_amdgcn_wmma_f32_16x16x4_f32(false, a2, false, bv, (short)0, acc2, false, false);
        acc3 = __builtin_amdgcn_wmma_f32_16x16x4_f32(false, a3, false, bv, (short)0, acc3, false, false);
    }

#pragma unroll
    for (int r = 0; r < 8; ++r) {
        const int row = r + 8 * hi;
        Q[(size_t)((m0 + 0) * 16 + row) * DETS + col] = acc0[r];
        Q[(size_t)((m0 + 1) * 16 + row) * DETS + col] = acc1[r];
        Q[(size_t)((m0 + 2) * 16 + row) * DETS + col] = acc2[r];
        Q[(size_t)((m0 + 3) * 16 + row) * DETS + col] = acc3[r];
    }
}

// -------------------------------------------------------------------
// Kernel 3: pixel-driven fan-beam backprojection.
// Block = one (batch, 16x16 pixel tile). For each of the 512 views the
// 736-float filtered row is staged in LDS (cooperative load), then every
// thread does the 1/U^2-weighted linear interpolation. cos/sin come from
// the precomputed table (uniform, L2-resident). Next row is prefetched.
// COEF is already folded into the filter matrix.
// -------------------------------------------------------------------
__global__ void __launch_bounds__(256)
fbp_backproject(const float* __restrict__ Q,    // [BATCH, VIEWS, DETS]
                const float* __restrict__ trig, // [VIEWS][2] = cos,sin
                float* __restrict__ out) {      // [BATCH, IMG_H, IMG_W]
    __shared__ float srow[DETS];

    const int b  = blockIdx.z;
    const int px = blockIdx.x * 16 + (threadIdx.x & 15);
    const int py = blockIdx.y * 16 + (threadIdx.x >> 4);

    const float X = ((float)px - (IMG_W - 1) * 0.5f) * D_IMG;
    const float Y = ((IMG_H - 1) * 0.5f - (float)py) * D_IMG;

    const float* Qb = Q + (size_t)b * VIEWS * DETS;
    float acc = 0.0f;

    for (int v = 0; v < VIEWS; ++v) {
        __syncthreads();                         // protect previous-view reads
        const float* row = Qb + (size_t)v * DETS;
#pragma unroll
        for (int i = threadIdx.x; i < DETS; i += 256) srow[i] = row[i];
        if (v + 1 < VIEWS)
            __builtin_prefetch(Qb + (size_t)(v + 1) * DETS + threadIdx.x, 0, 3);
        __syncthreads();

        const float cb = trig[2 * v + 0];
        const float sb = trig[2 * v + 1];
        const float U    = S2R - (X * cb + Y * sb);
        const float invU = __frcp_rn(U);
        const float u    = S2R * (Y * cb - X * sb) * invU;
        const float wgt  = (S2R * S2R) * invU * invU;
        const float idx  = u * (1.0f / VIRDET) + (DETS - 1) * 0.5f;

        if (idx >= 0.0f && idx <= (float)(DETS - 1)) {
            int i0 = (int)floorf(idx);
            i0 = i0 < 0 ? 0 : (i0 > DETS - 2 ? DETS - 2 : i0);
            float frac = idx - (float)i0;
            frac = frac < 0.0f ? 0.0f : (frac > 1.0f ? 1.0f : frac);
            const float v0 = srow[i0];
            const float v1 = srow[i0 + 1];
            acc += wgt * ((1.0f - frac) * v0 + frac * v1);
        }
    }

    out[((size_t)b * IMG_H + py) * IMG_W + px] = acc;
}

// -------------------------------------------------------------------
extern "C" void kernel_launch(void* const* d_in, const int* in_sizes, int n_in,
                              void* d_out, int out_size, void* d_ws, size_t ws_size,
                              hipStream_t stream) {
    const float* proj = (const float*)d_in[0];   // [4,1,512,736] fp32
    const float* filt = (const float*)d_in[1];   // [1471] fp32
    const float* w    = (const float*)d_in[2];   // [736] fp32
    float* out = (float*)d_out;                  // [4,1,256,256] fp32

    float* Ft   = (float*)d_ws;                              // 736*736
    float* Q    = Ft + (size_t)DETS * DETS;                  // 2048*736
    float* trig = Q + (size_t)BATCH * VIEWS * DETS;          // 1024

    // 1) fused filter matrix + per-view trig table
    fbp_build_tables<<<(DETS * DETS + 255) / 256, 256, 0, stream>>>(filt, w, Ft, trig);

    // 2) ramp filtering as fp32 WMMA GEMM: [2048x736] x [736x736]
    //    grid.x = 46 N-tiles, grid.y*8 waves * 4 M-tiles = 128 M-tiles
    fbp_filter_gemm<<<dim3(46, 4), 256, 0, stream>>>(proj, Ft, Q);

    // 3) backprojection: one block per (16x16 pixel tile, batch)
    fbp_backproject<<<dim3(IMG_W / 16, IMG_H / 16, BATCH), 256, 0, stream>>>(Q, trig, out);
}